// Upsample2d_1580547971799
// MI455X (gfx1250) — compile-verified
//
#include <hip/hip_runtime.h>

// upfirdn2d 2x upsample, separable [1,3,3,1]/4 per axis (overall /16).
// x: (8,64,224,224) f32 -> out: (8,64,448,448) f32.
// HBM-bound (~0.52 GB): TDM-staged LDS tiles, double-buffered across 8 planes
// per block to overlap tensor DMA with compute; NT b128 output stores.

typedef unsigned int u32x4 __attribute__((ext_vector_type(4)));
typedef int          i32x8 __attribute__((ext_vector_type(8)));
typedef int          i32x4 __attribute__((ext_vector_type(4)));
typedef float        f32x4 __attribute__((ext_vector_type(4)));
typedef float        f32x2 __attribute__((ext_vector_type(2)));

#define IN_W    224
#define IN_H    224
#define OUT_W   448
#define NPLANES 512          // 8 * 64
#define TILE_W  64           // input cols per block
#define TILE_H  16           // input rows per block
#define LDS_W   68           // LDS row stride in floats (66 used + slack)
#define LDS_H   18           // TILE_H + 2 halo rows
#define BUFSZ   (LDS_H * LDS_W)
#define NZ      8            // planes per block (double-buffered)
#define NTHREADS 512

__device__ __forceinline__ void tdm_load_tile(const float* x, unsigned laddr,
                                              int z, int r0, int c0,
                                              unsigned tile_d0, unsigned nrows)
{
#if defined(__AMDGCN__)
    const unsigned tdim0 = (unsigned)(IN_W - c0);   // valid cols from tile start (HW zero-fills beyond)
    unsigned long long gaddr = (unsigned long long)(uintptr_t)x
        + 4ull * ((unsigned long long)z * (IN_W * IN_H)
                  + (unsigned long long)r0 * IN_W
                  + (unsigned long long)c0);

    // D# group 0: count=1 | lds_addr | global_addr[56:0] | type=2
    u32x4 g0;
    g0[0] = 1u;
    g0[1] = laddr;
    g0[2] = (unsigned)(gaddr & 0xFFFFFFFFull);
    g0[3] = (unsigned)((gaddr >> 32) & 0x1FFFFFFull) | (2u << 30);

    // D# group 1: data_size=4B, iterate_enable; 1-row tiles so LDS stride stays LDS_W.
    i32x8 g1;
    g1[0] = (int)((2u << 16) | (1u << 19));            // data_size=2(4B) | iterate_enable
    g1[1] = (int)((tdim0 & 0xFFFFu) << 16);            // tensor_dim0[15:0]
    g1[2] = (int)((tdim0 >> 16) | (1u << 16));         // tensor_dim0[31:16] | tensor_dim1=1
    g1[3] = (int)(tile_d0 << 16);                      // tile_dim0 (row length incl. OOB zero-fill)
    g1[4] = 1;                                         // tile_dim1=1, tile_dim2=0
    g1[5] = IN_W;                                      // tensor_dim0_stride lo
    g1[6] = 0;
    g1[7] = 0;

    // D# group 2 (iterate mode): per-row address increments + row count.
    i32x4 g2;
    g2[0] = 0;
    g2[1] = LDS_W;                                     // lds_addr_increment (elems)
    g2[2] = IN_W;                                      // global_addr_increment (elems)
    g2[3] = (int)((nrows - 1u) << 16);                 // iterate_count

    i32x4 g3 = {0, 0, 0, 0};
    i32x8 gx = {0, 0, 0, 0, 0, 0, 0, 0};

    __builtin_amdgcn_tensor_load_to_lds(g0, g1, g2, g3, gx, 0);
#else
    (void)x; (void)laddr; (void)z; (void)r0; (void)c0; (void)tile_d0; (void)nrows;
#endif
}

__global__ __launch_bounds__(NTHREADS)
void upfirdn2d_up2_kernel(const float* __restrict__ x, float* __restrict__ out)
{
    __shared__ float smem[2 * BUFSZ];
    const int tid = threadIdx.x;
    const int bx  = blockIdx.x;              // 0..2   (W tiles)
    const int by  = blockIdx.y;              // 0..13  (H tiles)
    const int z0  = blockIdx.z * NZ;         // first of NZ planes

    // Tile geometry (uniform; same for every plane this block touches).
    const int gx0 = bx * TILE_W - 1;                    // leftmost col incl. halo (may be -1)
    const int gy0 = by * TILE_H - 1;                    // topmost row incl. halo (may be -1)
    const int c0  = gx0 < 0 ? 0 : gx0;
    const int r0  = gy0 < 0 ? 0 : gy0;
    const int ldsx = c0 - gx0;                          // 0 or 1
    const int ldsy = r0 - gy0;                          // 0 or 1
    int rend = gy0 + TILE_H + 1; if (rend > IN_H - 1) rend = IN_H - 1;
    const unsigned nrows   = (unsigned)(rend - r0 + 1);
    const unsigned tile_d0 = (unsigned)(TILE_W + 2 - ldsx);   // right OOB -> HW zero-fill
    const unsigned lds_off = 4u * (unsigned)(ldsy * LDS_W + ldsx);
    const unsigned lbase   = (unsigned)(uintptr_t)(&smem[0]);

    // Zero-fill only where the TDM never writes: left col (bx==0),
    // top halo row (by==0), bottom halo row (by==last).
    if (bx == 0 || by == 0 || by == (IN_H / TILE_H) - 1) {
        for (int i = tid; i < 2 * BUFSZ; i += NTHREADS) smem[i] = 0.0f;
        __syncthreads();
    }

    // Prologue: stage plane z0 into buffer 0.
    if (tid < 32) {
        tdm_load_tile(x, lbase + lds_off, z0, r0, c0, tile_d0, nrows);
#if defined(__AMDGCN__)
        __builtin_amdgcn_s_wait_tensorcnt(0);
#endif
    }
    __syncthreads();

    const int tx = tid & 31;                 // col pair within tile
    const int ty = tid >> 5;                 // row within tile (0..15)
    const int ocol = 4 * (32 * bx + tx);
    const int orow = 2 * (TILE_H * by + ty);
    const bool active = (ocol < OUT_W);
    const float s = 1.0f / 16.0f;

    for (int i = 0; i < NZ; ++i) {
        // Kick off next plane's DMA into the other buffer while we compute.
        if (i + 1 < NZ && tid < 32) {
            tdm_load_tile(x, lbase + (unsigned)(((i + 1) & 1) * BUFSZ * 4) + lds_off,
                          z0 + i + 1, r0, c0, tile_d0, nrows);
        }

        const float* sb  = &smem[(i & 1) * BUFSZ];
        const float* r0p = sb + (ty    ) * LDS_W + 2 * tx;
        const float* r1p = sb + (ty + 1) * LDS_W + 2 * tx;
        const float* r2p = sb + (ty + 2) * LDS_W + 2 * tx;

        f32x4 hm, hc, hp;
        {
            f32x2 a = *(const f32x2*)(r0p), b = *(const f32x2*)(r0p + 2);
            hm[0] = a[0] + 3.0f*a[1];  hm[1] = 3.0f*a[1] + b[0];
            hm[2] = a[1] + 3.0f*b[0];  hm[3] = 3.0f*b[0] + b[1];
        }
        {
            f32x2 a = *(const f32x2*)(r1p), b = *(const f32x2*)(r1p + 2);
            hc[0] = a[0] + 3.0f*a[1];  hc[1] = 3.0f*a[1] + b[0];
            hc[2] = a[1] + 3.0f*b[0];  hc[3] = 3.0f*b[0] + b[1];
        }
        {
            f32x2 a = *(const f32x2*)(r2p), b = *(const f32x2*)(r2p + 2);
            hp[0] = a[0] + 3.0f*a[1];  hp[1] = 3.0f*a[1] + b[0];
            hp[2] = a[1] + 3.0f*b[0];  hp[3] = 3.0f*b[0] + b[1];
        }

        f32x4 top, bot;
        top[0] = (hm[0] + 3.0f*hc[0]) * s;  top[1] = (hm[1] + 3.0f*hc[1]) * s;
        top[2] = (hm[2] + 3.0f*hc[2]) * s;  top[3] = (hm[3] + 3.0f*hc[3]) * s;
        bot[0] = (3.0f*hc[0] + hp[0]) * s;  bot[1] = (3.0f*hc[1] + hp[1]) * s;
        bot[2] = (3.0f*hc[2] + hp[2]) * s;  bot[3] = (3.0f*hc[3] + hp[3]) * s;

        if (active) {
            size_t base = ((size_t)(z0 + i) * OUT_W + (size_t)orow) * OUT_W + (size_t)ocol;
            __builtin_nontemporal_store(top, (f32x4*)(&out[base]));          // b128 NT
            __builtin_nontemporal_store(bot, (f32x4*)(&out[base + OUT_W])); // b128 NT
        }

        // Drain the async tensor load, then make the new buffer visible to all waves.
        if (tid < 32) {
#if defined(__AMDGCN__)
            __builtin_amdgcn_s_wait_tensorcnt(0);
#endif
        }
        __syncthreads();
    }
}

extern "C" void kernel_launch(void* const* d_in, const int* in_sizes, int n_in,
                              void* d_out, int out_size, void* d_ws, size_t ws_size,
                              hipStream_t stream) {
    (void)in_sizes; (void)n_in; (void)d_ws; (void)ws_size; (void)out_size;
    const float* x   = (const float*)d_in[0];   // (8,64,224,224) f32
    float*       out = (float*)d_out;           // (8,64,448,448) f32
    dim3 block(NTHREADS);
    dim3 grid(IN_W / TILE_W, IN_H / TILE_H, NPLANES / NZ);   // (3, 14, 64)
    upfirdn2d_up2_kernel<<<grid, block, 0, stream>>>(x, out);
}